// GCNEdgeDot_32701880992039
// MI455X (gfx1250) — compile-verified
//
#include <hip/hip_runtime.h>
#include <hip/hip_bf16.h>

// ---------------------------------------------------------------------------
// GCN 2-layer + edge-dot scoring for MI455X (gfx1250, wave32, WMMA).
//   agg = segment_sum(vals * X[col]) over rows  (SpMM via f32 atomics, L2-resident)
//   H1  = relu(agg @ Wp1 + X @ Ws1 + b)        (fp32 WMMA 16x16x4)
//   agg = segment_sum(vals * H1[col])
//   H2  = relu(agg @ Wp2 + H1 @ Ws2 + b)       (fp32 WMMA 16x16x4)
//   out = sigmoid(<H2[row], H2[col]>)          (wave-per-edge shuffle reduce)
// ---------------------------------------------------------------------------

typedef float v2f __attribute__((ext_vector_type(2)));
typedef float v8f __attribute__((ext_vector_type(8)));

#define N_NODES 50000
#define N_EDGES 800000
#define D_IN    128
#define D_H1    128
#define D_H2    64

// ---------------------------------------------------------------------------
// SpMM: agg[row[e], :] += vals[e] * Xin[col[e], :]   (D = 128 floats per row)
// One wave per edge; lane c handles float4 chunk c (32 chunks * 16B = 512B row).
// Row gather is fully coalesced; scatter uses global_atomic_add_f32 (L2).
// ---------------------------------------------------------------------------
__global__ __launch_bounds__(256) void spmm_atomic_kernel(
    const float* __restrict__ Xin, const int* __restrict__ erow,
    const int* __restrict__ ecol, const float* __restrict__ evals,
    float* __restrict__ agg, int nEdges) {
  int t = blockIdx.x * blockDim.x + threadIdx.x;
  int e = t >> 5;          // wave handles one edge
  int c = t & 31;          // float4 chunk within the 128-float row
  if (e >= nEdges) return;
  int r = erow[e];
  int j = ecol[e];
  float v = evals[e];
  const float4 x = reinterpret_cast<const float4*>(Xin + (size_t)j * D_IN)[c];
  float* dst = agg + (size_t)r * D_IN + c * 4;
  atomicAdd(dst + 0, v * x.x);
  atomicAdd(dst + 1, v * x.y);
  atomicAdd(dst + 2, v * x.z);
  atomicAdd(dst + 3, v * x.w);
}

// ---------------------------------------------------------------------------
// Fused dual GEMM + bias + ReLU:
//   out[M, N] = relu(A1 @ W1 + A2 @ W2 + (b1 + b2))     K = 128, N = NT*16
// One wave owns one 16x16 output tile (block covers 16 rows x N cols).
// grid.x = M/16 (M = 50000 = 3125*16, exact -> EXEC all ones, required by WMMA).
//
// V_WMMA_F32_16X16X4_F32 operand layout (ISA 7.12.2):
//   A (16x4): lane L -> M = L&15, K = {2*(L>>4), 2*(L>>4)+1}   (v2f)
//   B (4x16): lane L -> N = L&15, K = {2*(L>>4), 2*(L>>4)+1}   (v2f)
//   C/D     : vgpr r, lane L -> M = r + 8*(L>>4), N = L&15     (v8f)
// ---------------------------------------------------------------------------
template <int KDIM, int NT>
__global__ __launch_bounds__(NT * 32) void gemm2_bias_relu_kernel(
    const float* __restrict__ A1, const float* __restrict__ A2,
    const float* __restrict__ W1, const float* __restrict__ W2,
    const float* __restrict__ b1, const float* __restrict__ b2,
    float* __restrict__ out) {
  constexpr int N = NT * 16;
  const int lane = threadIdx.x & 31;
  const int wave = threadIdx.x >> 5;
  const int n0 = wave * 16;            // this wave's 16-column tile
  const int m0 = blockIdx.x * 16;      // this block's 16-row tile
  const int m = lane & 15;
  const int hi = lane >> 4;            // 0: K pair {0,1}; 1: K pair {2,3}
  const int n = lane & 15;

  const float* pA1 = A1 + (size_t)(m0 + m) * KDIM + 2 * hi;
  const float* pA2 = A2 + (size_t)(m0 + m) * KDIM + 2 * hi;
  const float* pW1 = W1 + n0 + n;
  const float* pW2 = W2 + n0 + n;

  v8f acc = {};
#pragma unroll
  for (int k = 0; k < KDIM; k += 4) {
    const int kr = k + 2 * hi;
    v2f a1 = *reinterpret_cast<const v2f*>(pA1 + k);
    v2f a2 = *reinterpret_cast<const v2f*>(pA2 + k);
    v2f w1, w2;
    w1.x = pW1[(size_t)kr * N];
    w1.y = pW1[(size_t)(kr + 1) * N];
    w2.x = pW2[(size_t)kr * N];
    w2.y = pW2[(size_t)(kr + 1) * N];
    acc = __builtin_amdgcn_wmma_f32_16x16x4_f32(false, a1, false, w1,
                                                (short)0, acc, false, false);
    acc = __builtin_amdgcn_wmma_f32_16x16x4_f32(false, a2, false, w2,
                                                (short)0, acc, false, false);
  }

  const float bias = b1[n0 + n] + b2[n0 + n];
#pragma unroll
  for (int r = 0; r < 8; ++r) {
    const int row = m0 + r + 8 * hi;
    float v = acc[r] + bias;
    out[(size_t)row * N + n0 + n] = v > 0.0f ? v : 0.0f;
  }
}

// ---------------------------------------------------------------------------
// Edge scoring: out[e] = sigmoid(dot(H2[row[e]], H2[col[e]]))  (64-dim rows)
// One wave per edge: lane loads float2 from each endpoint (coalesced 256B),
// 5-step xor-shuffle reduction, lane 0 writes sigmoid.
// ---------------------------------------------------------------------------
__global__ __launch_bounds__(256) void edge_score_kernel(
    const float* __restrict__ H2, const int* __restrict__ erow,
    const int* __restrict__ ecol, float* __restrict__ out, int nEdges) {
  int t = blockIdx.x * blockDim.x + threadIdx.x;
  int e = t >> 5;
  int lane = t & 31;
  if (e >= nEdges) return;
  int r = erow[e];
  int c = ecol[e];
  float2 a = reinterpret_cast<const float2*>(H2 + (size_t)r * D_H2)[lane];
  float2 b = reinterpret_cast<const float2*>(H2 + (size_t)c * D_H2)[lane];
  float p = a.x * b.x + a.y * b.y;
#pragma unroll
  for (int off = 16; off > 0; off >>= 1) p += __shfl_xor(p, off, 32);
  if (lane == 0) out[e] = 1.0f / (1.0f + __expf(-p));
}

// ---------------------------------------------------------------------------
// Host orchestration (graph-capture safe: only async ops on `stream`).
// ---------------------------------------------------------------------------
extern "C" void kernel_launch(void* const* d_in, const int* in_sizes, int n_in,
                              void* d_out, int out_size, void* d_ws,
                              size_t ws_size, hipStream_t stream) {
  const float* X     = (const float*)d_in[0];
  const int*   erow  = (const int*)d_in[1];
  const int*   ecol  = (const int*)d_in[2];
  const float* evals = (const float*)d_in[3];
  const float* Wp1   = (const float*)d_in[4];
  const float* bp1   = (const float*)d_in[5];
  const float* Ws1   = (const float*)d_in[6];
  const float* bs1   = (const float*)d_in[7];
  const float* Wp2   = (const float*)d_in[8];
  const float* bp2   = (const float*)d_in[9];
  const float* Ws2   = (const float*)d_in[10];
  const float* bs2   = (const float*)d_in[11];
  float* out = (float*)d_out;

  const int nEdges = in_sizes[1];           // 800000
  const int M      = in_sizes[0] / D_IN;    // 50000 (divisible by 16)

  // Workspace layout: agg (reused by both layers) | H1 | H2
  const size_t aggBytes = (size_t)M * D_IN * sizeof(float);   // 25.6 MB
  const size_t h1Bytes  = (size_t)M * D_H1 * sizeof(float);   // 25.6 MB
  char* ws   = (char*)d_ws;
  float* agg = (float*)(ws);
  float* H1  = (float*)(ws + aggBytes);
  float* H2  = (float*)(ws + aggBytes + h1Bytes);

  const int spmmBlocks = (nEdges * 32 + 255) / 256;  // wave per edge
  const int gemmBlocks = M / 16;                      // 3125, exact
  const int edgeBlocks = (nEdges * 32 + 255) / 256;

  // ---- Layer 1 ----
  hipMemsetAsync(agg, 0, aggBytes, stream);
  spmm_atomic_kernel<<<spmmBlocks, 256, 0, stream>>>(X, erow, ecol, evals,
                                                     agg, nEdges);
  gemm2_bias_relu_kernel<D_IN, D_H1 / 16>
      <<<gemmBlocks, (D_H1 / 16) * 32, 0, stream>>>(agg, X, Wp1, Ws1, bp1,
                                                    bs1, H1);

  // ---- Layer 2 ----
  hipMemsetAsync(agg, 0, aggBytes, stream);
  spmm_atomic_kernel<<<spmmBlocks, 256, 0, stream>>>(H1, erow, ecol, evals,
                                                     agg, nEdges);
  gemm2_bias_relu_kernel<D_H1, D_H2 / 16>
      <<<gemmBlocks, (D_H2 / 16) * 32, 0, stream>>>(agg, H1, Wp2, Ws2, bp2,
                                                    bs2, H2);

  // ---- Edge scores ----
  edge_score_kernel<<<edgeBlocks, 256, 0, stream>>>(H2, erow, ecol, out,
                                                    nEdges);
}